// PointNetfeat_54657753809449
// MI455X (gfx1250) — compile-verified
//
#include <hip/hip_runtime.h>
#include <math.h>

#define EPSBN 1e-5f
#define KNN 5
#define NPTS 2048
#define BATCH 8

typedef float v2f __attribute__((ext_vector_type(2)));
typedef float v8f __attribute__((ext_vector_type(8)));

__device__ inline void atomicMaxF(float* addr, float val) {
  int* ai = (int*)addr;
  int old = *ai;
  while (__int_as_float(old) < val) {
    int prev = atomicCAS(ai, old, __float_as_int(val));
    if (prev == old) break;
    old = prev;
  }
}

__device__ inline v8f wmma_f32(v2f a, v2f b, v8f c) {
  return __builtin_amdgcn_wmma_f32_16x16x4_f32(false, a, false, b, (short)0, c, false, false);
}

// ---------------- utility ----------------
__global__ void k_fill(float* p, long long n, float v) {
  long long i = blockIdx.x * (long long)blockDim.x + threadIdx.x;
  long long stride = (long long)gridDim.x * blockDim.x;
  for (; i < n; i += stride) p[i] = v;
}

__global__ void k_transpose_x(const float* x, float* xt) {
  int i = blockIdx.x * blockDim.x + threadIdx.x; // B*N*5
  if (i >= BATCH * NPTS * 5) return;
  int c = i % 5; int n = (i / 5) % NPTS; int b = i / (5 * NPTS);
  xt[i] = x[(b * 5 + c) * NPTS + n];
}

// ---------------- KNN top-5 (maximize pd = 2 x.y - |x|^2 - |y|^2) ----------------
__global__ __launch_bounds__(256) void k_knn(const float* xf, int C, int* out_idx) {
  __shared__ float ctr[128];
  __shared__ float tv[256 * KNN];
  __shared__ int   ti[256 * KNN];
  int row = blockIdx.x; int b = row / NPTS; int n = row % NPTS;
  int t = threadIdx.x;
  const float* base = xf + (long long)b * NPTS * C;
  for (int c = t; c < C; c += 256) ctr[c] = base[(long long)n * C + c];
  __syncthreads();
  float d2n = 0.f;
  for (int c = 0; c < C; ++c) d2n += ctr[c] * ctr[c];
  float bv[KNN]; int bi[KNN];
  for (int s = 0; s < KNN; ++s) { bv[s] = -INFINITY; bi[s] = 0x7fffffff; }
  for (int m = t; m < NPTS; m += 256) {
    const float* pm = base + (long long)m * C;
    float dot = 0.f, d2m = 0.f;
    for (int c = 0; c < C; ++c) { float v = pm[c]; dot += ctr[c] * v; d2m += v * v; }
    float pd = 2.f * dot - d2n - d2m;
    if (pd > bv[KNN - 1]) {
      int s = KNN - 1;
      while (s > 0 && pd > bv[s - 1]) { bv[s] = bv[s - 1]; bi[s] = bi[s - 1]; --s; }
      bv[s] = pd; bi[s] = m;
    }
  }
  for (int s = 0; s < KNN; ++s) { tv[t * KNN + s] = bv[s]; ti[t * KNN + s] = bi[s]; }
  __syncthreads();
  if (t == 0) {
    for (int s = 0; s < KNN; ++s) {
      float best = -INFINITY; int bidx = 0x7fffffff; int bpos = 0;
      for (int p = 0; p < 256 * KNN; ++p) {
        float v = tv[p]; int id = ti[p];
        if (v > best || (v == best && id < bidx)) { best = v; bidx = id; bpos = p; }
      }
      tv[bpos] = -INFINITY;
      out_idx[(long long)row * KNN + s] = bidx;
    }
  }
}

// ---------------- edge feature build: concat(nb-ctr, ctr), zero-padded to ld ----------------
__global__ void k_build_feat(const float* xf, const int* idx, int C, int ld, float* feat) {
  long long total = (long long)BATCH * NPTS * KNN * ld;
  long long i = blockIdx.x * (long long)blockDim.x + threadIdx.x;
  long long stride = (long long)gridDim.x * blockDim.x;
  for (; i < total; i += stride) {
    int c = (int)(i % ld); long long r = i / ld;
    long long bn = r / KNN;
    int n = (int)(bn % NPTS); int b = (int)(bn / NPTS);
    float v = 0.f;
    if (c < 2 * C) {
      const float* base = xf + (long long)b * NPTS * C;
      if (c < C) {
        int j = idx[r];
        v = base[(long long)j * C + c] - base[(long long)n * C + c];
      } else {
        v = base[(long long)n * C + (c - C)];
      }
    }
    feat[i] = v;
  }
}

// ---------------- stats: column sums and second-moment S = A^T A ----------------
__global__ void k_colsum(const float* A, long long M, int C, int ld, long long chunk, float* mu) {
  int c = blockIdx.x * 64 + threadIdx.x;
  if (c >= C) return;
  long long r0 = (long long)blockIdx.y * chunk;
  long long r1 = r0 + chunk; if (r1 > M) r1 = M;
  float acc = 0.f;
  for (long long r = r0; r < r1; ++r) acc += A[r * ld + c];
  atomicAdd(&mu[c], acc);
}

__global__ __launch_bounds__(256) void k_smat(const float* A, long long M, int C, int ld,
                                              long long chunk, float* S) {
  __shared__ float Ai[32][16], Aj[32][16];
  int i0 = blockIdx.x * 16, j0 = blockIdx.y * 16;
  long long r0 = (long long)blockIdx.z * chunk;
  long long r1 = r0 + chunk; if (r1 > M) r1 = M;
  int t = threadIdx.x;
  int ii = t / 16, jj = t % 16;
  float acc = 0.f;
  for (long long rs = r0; rs < r1; rs += 32) {
    long long nrl = r1 - rs; int nr = (nrl > 32) ? 32 : (int)nrl;
    for (int e = t; e < 32 * 16; e += 256) {
      int rl = e / 16, cl = e % 16;
      float vi = 0.f, vj = 0.f;
      if (rl < nr) {
        long long r = rs + rl;
        if (i0 + cl < C) vi = A[r * ld + i0 + cl];
        if (j0 + cl < C) vj = A[r * ld + j0 + cl];
      }
      Ai[rl][cl] = vi; Aj[rl][cl] = vj;
    }
    __syncthreads();
    for (int rl = 0; rl < 32; ++rl) acc += Ai[rl][ii] * Aj[rl][jj];
    __syncthreads();
  }
  if (i0 + ii < C && j0 + jj < C) atomicAdd(&S[(i0 + ii) * C + j0 + jj], acc);
}

// ---------------- BN coeffs: scale = g*rsqrt(var+eps), shift = beta - scale*mean ----------------
__global__ __launch_bounds__(256) void k_bncoef(const float* W, int Cin, const float* g,
                                                const float* beta, const float* mu,
                                                const float* S, float invM,
                                                float* scale, float* shift) {
  int o = blockIdx.x;
  const float* w = W + (long long)o * Cin;
  __shared__ float red[256];
  int t = threadIdx.x;
  float acc = 0.f;
  int CC = Cin * Cin;
  for (int p = t; p < CC; p += 256) {
    int i = p / Cin, j = p % Cin;
    acc += w[i] * S[p] * w[j];
  }
  red[t] = acc; __syncthreads();
  for (int s = 128; s > 0; s >>= 1) { if (t < s) red[t] += red[t + s]; __syncthreads(); }
  float wSw = red[0];
  __syncthreads();
  float acc2 = 0.f;
  for (int i = t; i < Cin; i += 256) acc2 += w[i] * mu[i];
  red[t] = acc2; __syncthreads();
  for (int s = 128; s > 0; s >>= 1) { if (t < s) red[t] += red[t + s]; __syncthreads(); }
  if (t == 0) {
    float m = red[0] * invM;
    float var = wSw * invM - m * m;
    float a = g[o] * rsqrtf(var + EPSBN);
    scale[o] = a;
    shift[o] = beta[o] - a * m;
  }
}

// ---------------- WMMA GEMM + BN + activation (STN convs): 32-col tile, 2 acc/wave -------------
// Guard-free inner loop: B staged in LDS zero-padded to Kpad; A overshoot (conv1 Kdim=5)
// reads in-bounds garbage multiplied by zero-padded B. Row strides padded (34) so the two
// half-waves (rows k / k+2) hit disjoint LDS banks.
__global__ __launch_bounds__(256) void k_gemm_act(const float* A, int ldA, const float* W,
                                                  int Kdim, const float* scale,
                                                  const float* shift, float slope,
                                                  float* out, int ldOut) {
  __shared__ float Wlds[128][34];
  int wave = threadIdx.x >> 5;
  int lane = threadIdx.x & 31;
  int row0 = (blockIdx.x * 8 + wave) * 16;
  int col0 = blockIdx.y * 32;
  int mn = lane & 15;
  int part = lane >> 4;
  int kb = part * 2;
  int Kpad = (Kdim + 3) & ~3;
  for (int e = threadIdx.x; e < Kpad * 32; e += 256) {
    int k = e >> 5, c = e & 31;
    Wlds[k][c] = (k < Kdim) ? W[(long long)(col0 + c) * Kdim + k] : 0.f;
  }
  __syncthreads();
  v8f acc0 = {0.f, 0.f, 0.f, 0.f, 0.f, 0.f, 0.f, 0.f};
  v8f acc1 = acc0;
  const float* ap = A + (long long)(row0 + mn) * ldA + kb;
#pragma unroll 4
  for (int k = 0; k < Kpad; k += 4) {
    v2f a, b0, b1;
    a.x = ap[k];
    a.y = ap[k + 1];
    b0.x = Wlds[k + kb][mn];
    b0.y = Wlds[k + kb + 1][mn];
    b1.x = Wlds[k + kb][mn + 16];
    b1.y = Wlds[k + kb + 1][mn + 16];
    acc0 = wmma_f32(a, b0, acc0);
    acc1 = wmma_f32(a, b1, acc1);
  }
  for (int v = 0; v < 8; ++v) {
    int r = v + 8 * part;
    float y0 = acc0[v] * scale[col0 + mn] + shift[col0 + mn];
    float y1 = acc1[v] * scale[col0 + mn + 16] + shift[col0 + mn + 16];
    y0 = (y0 > 0.f) ? y0 : slope * y0;
    y1 = (y1 > 0.f) ? y1 : slope * y1;
    float* orow = out + (long long)(row0 + r) * ldOut + col0;
    orow[mn] = y0;
    orow[mn + 16] = y1;
  }
}

// ---------------- WMMA GEMM + BN + leaky + max over KNN=5: 5 waves, 32-col tile ---------------
// A (edge features) is zero-padded to ld = Kpad by k_build_feat, so loop is guard-free.
__global__ __launch_bounds__(160) void k_gemm_edge(const float* A, int ldA, const float* W,
                                                   int Kdim, int Ocols, const float* scale,
                                                   const float* shift, float slope,
                                                   float* out, int ldOut, int atomicMode) {
  __shared__ float Wlds[256][34];
  __shared__ float tile[80][33];
  int wave = threadIdx.x >> 5;   // 0..4
  int lane = threadIdx.x & 31;
  int row0 = blockIdx.x * 80 + wave * 16;
  int col0 = blockIdx.y * 32;
  int mn = lane & 15;
  int part = lane >> 4;
  int kb = part * 2;
  int Kpad = (Kdim + 3) & ~3;
  for (int e = threadIdx.x; e < Kpad * 32; e += 160) {
    int k = e >> 5, c = e & 31;
    Wlds[k][c] = (k < Kdim) ? W[(long long)(col0 + c) * Kdim + k] : 0.f;
  }
  __syncthreads();
  v8f acc0 = {0.f, 0.f, 0.f, 0.f, 0.f, 0.f, 0.f, 0.f};
  v8f acc1 = acc0;
  const float* ap = A + (long long)(row0 + mn) * ldA + kb;
#pragma unroll 4
  for (int k = 0; k < Kpad; k += 4) {
    v2f a, b0, b1;
    a.x = ap[k];
    a.y = ap[k + 1];
    b0.x = Wlds[k + kb][mn];
    b0.y = Wlds[k + kb + 1][mn];
    b1.x = Wlds[k + kb][mn + 16];
    b1.y = Wlds[k + kb + 1][mn + 16];
    acc0 = wmma_f32(a, b0, acc0);
    acc1 = wmma_f32(a, b1, acc1);
  }
  for (int v = 0; v < 8; ++v) {
    int r = v + 8 * part;
    float y0 = acc0[v] * scale[col0 + mn] + shift[col0 + mn];
    float y1 = acc1[v] * scale[col0 + mn + 16] + shift[col0 + mn + 16];
    y0 = (y0 > 0.f) ? y0 : slope * y0;
    y1 = (y1 > 0.f) ? y1 : slope * y1;
    tile[wave * 16 + r][mn] = y0;
    tile[wave * 16 + r][mn + 16] = y1;
  }
  __syncthreads();
  for (int p = threadIdx.x; p < 512; p += 160) {
    int g = p / 32, c = p % 32;
    float mx = tile[g * 5][c];
    for (int j = 1; j < 5; ++j) mx = fmaxf(mx, tile[g * 5 + j][c]);
    long long grp = (long long)blockIdx.x * 16 + g;   // global b*N+n
    if (!atomicMode) {
      out[grp * ldOut + col0 + c] = mx;
    } else {
      int b = (int)(grp / NPTS);
      atomicMaxF(&out[(long long)b * Ocols + col0 + c], mx);
    }
  }
}

// ---------------- small kernels: pooling, FC, BN over 8 rows, trans build, transform -------------
__global__ void k_rowmax(const float* h, int C, float* pooled) {
  int i = blockIdx.x * blockDim.x + threadIdx.x;
  if (i >= BATCH * C) return;
  int b = i / C, c = i % C;
  const float* p = h + (long long)b * NPTS * C + c;
  float mx = -INFINITY;
  for (int n = 0; n < NPTS; ++n) mx = fmaxf(mx, p[(long long)n * C]);
  pooled[i] = mx;
}

__global__ void k_fc(const float* h, int Kd, const float* W, const float* bias, int O, float* z) {
  int i = blockIdx.x * blockDim.x + threadIdx.x;
  if (i >= BATCH * O) return;
  int r = i / O, o = i % O;
  const float* hr = h + (long long)r * Kd;
  const float* wo = W + (long long)o * Kd;
  float acc = bias[o];
  for (int k = 0; k < Kd; ++k) acc += hr[k] * wo[k];
  z[i] = acc;
}

__global__ void k_bnrows(float* z, int O, const float* g, const float* beta) {
  int o = blockIdx.x * blockDim.x + threadIdx.x;
  if (o >= O) return;
  float m = 0.f;
  for (int r = 0; r < BATCH; ++r) m += z[r * O + o];
  m /= (float)BATCH;
  float v = 0.f;
  for (int r = 0; r < BATCH; ++r) { float d = z[r * O + o] - m; v += d * d; }
  v /= (float)BATCH;
  float a = g[o] * rsqrtf(v + EPSBN);
  for (int r = 0; r < BATCH; ++r) {
    float y = a * (z[r * O + o] - m) + beta[o];
    z[r * O + o] = (y > 0.f) ? y : 0.f;
  }
}

__global__ void k_mktrans(const float* z, int k, float* T) {
  int i = blockIdx.x * blockDim.x + threadIdx.x;
  if (i >= BATCH * k * k) return;
  int rem = i % (k * k);
  int ii = rem / k, jj = rem % k;
  T[i] = z[i] + ((ii == jj) ? 1.f : 0.f);
}

__global__ void k_transform(float* x1, const float* T) {
  int i = blockIdx.x * blockDim.x + threadIdx.x; // per (b,n)
  if (i >= BATCH * NPTS) return;
  int b = i / NPTS;
  float row[64];
  float* xr = x1 + (long long)i * 64;
  for (int c = 0; c < 64; ++c) row[c] = xr[c];
  const float* Tb = T + b * 64 * 64;
  for (int d = 0; d < 64; ++d) {
    float acc = 0.f;
    for (int c = 0; c < 64; ++c) acc += row[c] * Tb[c * 64 + d];
    xr[d] = acc;
  }
}

__global__ void k_assemble(const float* x1, const float* x2, const float* x33,
                           const float* x4, const float* trans, const float* tf,
                           float* out) {
  const long long outMain = (long long)BATCH * 1280 * NPTS;
  const long long transSz = BATCH * 25;
  const long long tfSz = BATCH * 64 * 64;
  long long total = outMain + transSz + tfSz;
  long long i = blockIdx.x * (long long)blockDim.x + threadIdx.x;
  long long stride = (long long)gridDim.x * blockDim.x;
  for (; i < total; i += stride) {
    float v;
    if (i < outMain) {
      int n = (int)(i % NPTS);
      long long rc = i / NPTS;
      int c = (int)(rc % 1280);
      int b = (int)(rc / 1280);
      long long bn = (long long)b * NPTS + n;
      if (c < 64)        v = x1[bn * 64 + c];
      else if (c < 128)  v = x2[bn * 64 + (c - 64)];
      else if (c < 256)  v = x33[bn * 128 + (c - 128)];
      else               v = x4[b * 1024 + (c - 256)];
      out[i] = v;
    } else if (i < outMain + transSz) {
      out[i] = trans[i - outMain];
    } else {
      out[i] = tf[i - outMain - transSz];
    }
  }
}

// ---------------- host ----------------
extern "C" void kernel_launch(void* const* d_in, const int* in_sizes, int n_in,
                              void* d_out, int out_size, void* d_ws, size_t ws_size,
                              hipStream_t stream) {
  (void)in_sizes; (void)n_in; (void)out_size; (void)ws_size;
  const float* x = (const float*)d_in[0];
  auto P = [&](int i) { return (const float*)d_in[i]; };

  float* w = (float*)d_ws;
  size_t off = 0;
  auto alloc = [&](size_t n) { float* p = w + off; off += n; return p; };
  float* xt     = alloc((size_t)BATCH * NPTS * 5);
  float* x1     = alloc((size_t)BATCH * NPTS * 64);
  float* x2     = alloc((size_t)BATCH * NPTS * 64);
  float* x33    = alloc((size_t)BATCH * NPTS * 128);
  float* x4     = alloc(BATCH * 1024);
  float* trans  = alloc(BATCH * 25);
  float* tf     = alloc(BATCH * 64 * 64);
  int*   idx    = (int*)alloc((size_t)BATCH * NPTS * KNN);
  float* pooled = alloc(BATCH * 1024);
  float* fcb1   = alloc(BATCH * 512);
  float* fcb2   = alloc(BATCH * 256);
  float* fcb3   = alloc(BATCH * 4096);
  float* mu     = alloc(256);
  float* Sm     = alloc(256 * 256);
  float* scl    = alloc(1024);
  float* shf    = alloc(1024);
  float* bufBig = alloc((size_t)BATCH * NPTS * KNN * 256);
  float* bufP1  = alloc((size_t)BATCH * NPTS * 128);
  float* bufP2  = alloc((size_t)BATCH * NPTS * 128);

  auto fill = [&](float* p, long long n, float v) {
    long long b = (n + 255) / 256; if (b > 4096) b = 4096;
    k_fill<<<dim3((unsigned)b), dim3(256), 0, stream>>>(p, n, v);
  };
  auto stats = [&](const float* A, long long M, int C, int ld) {
    fill(mu, C, 0.f); fill(Sm, (long long)C * C, 0.f);
    long long chunk = 2048;
    int nch = (int)((M + chunk - 1) / chunk);
    k_colsum<<<dim3((C + 63) / 64, nch), dim3(64), 0, stream>>>(A, M, C, ld, chunk, mu);
    k_smat<<<dim3((C + 15) / 16, (C + 15) / 16, nch), dim3(256), 0, stream>>>(A, M, C, ld, chunk, Sm);
  };
  auto coeff = [&](const float* Wt, int Cin, const float* g, const float* beta, long long M, int O) {
    k_bncoef<<<dim3(O), dim3(256), 0, stream>>>(Wt, Cin, g, beta, mu, Sm, 1.f / (float)M, scl, shf);
  };
  auto gemm = [&](const float* A, int ldA, const float* Wt, int Kd, long long M, int O,
                  float slope, float* outp) {
    k_gemm_act<<<dim3((unsigned)(M / 128), O / 32), dim3(256), 0, stream>>>(
        A, ldA, Wt, Kd, scl, shf, slope, outp, O);
  };
  auto run_stn = [&](const float* inp, int Cin0, int pbase, float* transOut, int kdim) {
    const int convO[4] = {64, 64, 128, 1024};
    float* outs[4] = {bufP1, bufP2, bufP1, bufBig};
    const float* curA = inp; int curC = Cin0;
    long long M = (long long)BATCH * NPTS;
    for (int j = 0; j < 4; ++j) {
      const float* Wt = P(pbase + 4 * j + 0);
      const float* g  = P(pbase + 4 * j + 2);
      const float* be = P(pbase + 4 * j + 3);
      stats(curA, M, curC, curC);
      coeff(Wt, curC, g, be, M, convO[j]);
      gemm(curA, curC, Wt, curC, M, convO[j], 0.f, outs[j]);
      curA = outs[j]; curC = convO[j];
    }
    k_rowmax<<<dim3((BATCH * 1024 + 255) / 256), dim3(256), 0, stream>>>(bufBig, 1024, pooled);
    int fb = pbase + 16;
    k_fc<<<dim3((BATCH * 512 + 255) / 256), dim3(256), 0, stream>>>(pooled, 1024, P(fb + 0), P(fb + 1), 512, fcb1);
    k_bnrows<<<dim3(2), dim3(256), 0, stream>>>(fcb1, 512, P(fb + 2), P(fb + 3));
    k_fc<<<dim3((BATCH * 256 + 255) / 256), dim3(256), 0, stream>>>(fcb1, 512, P(fb + 4), P(fb + 5), 256, fcb2);
    k_bnrows<<<dim3(1), dim3(256), 0, stream>>>(fcb2, 256, P(fb + 6), P(fb + 7));
    int kk = kdim * kdim;
    k_fc<<<dim3((BATCH * kk + 255) / 256), dim3(256), 0, stream>>>(fcb2, 256, P(fb + 8), P(fb + 9), kk, fcb3);
    k_mktrans<<<dim3((BATCH * kk + 255) / 256), dim3(256), 0, stream>>>(fcb3, kdim, transOut);
  };
  auto run_edge = [&](const float* inp, int C, int ebase, int O, float* outp, int atomicMode) {
    k_knn<<<dim3(BATCH * NPTS), dim3(256), 0, stream>>>(inp, C, idx);
    int ld = ((2 * C + 3) / 4) * 4;
    long long M = (long long)BATCH * NPTS * KNN;
    long long total = M * ld;
    long long b = (total + 255) / 256; if (b > 8192) b = 8192;
    k_build_feat<<<dim3((unsigned)b), dim3(256), 0, stream>>>(inp, idx, C, ld, bufBig);
    stats(bufBig, M, 2 * C, ld);
    coeff(P(ebase + 0), 2 * C, P(ebase + 1), P(ebase + 2), M, O);
    k_gemm_edge<<<dim3(1024, O / 32), dim3(160), 0, stream>>>(
        bufBig, ld, P(ebase + 0), 2 * C, O, scl, shf, 0.2f, outp, O, atomicMode);
  };

  // pipeline
  k_transpose_x<<<dim3((BATCH * NPTS * 5 + 255) / 256), dim3(256), 0, stream>>>(x, xt);
  run_stn(xt, 5, 1, trans, 5);                 // STN (k=5): params [1..28]
  run_edge(xt, 5, 57, 64, x1, 0);              // edge0: params [57..59]
  run_stn(x1, 64, 29, tf, 64);                 // fSTN: params [29..56]
  k_transform<<<dim3((BATCH * NPTS + 255) / 256), dim3(256), 0, stream>>>(x1, tf);
  run_edge(x1, 64, 60, 64, x2, 0);             // edge1
  run_edge(x2, 64, 63, 128, x33, 0);           // edge2
  fill(x4, BATCH * 1024, -INFINITY);
  run_edge(x33, 128, 66, 1024, x4, 1);         // edge3 -> atomic max over N
  {
    long long total = (long long)BATCH * 1280 * NPTS + BATCH * 25 + BATCH * 64 * 64;
    long long b = (total + 255) / 256; if (b > 8192) b = 8192;
    k_assemble<<<dim3((unsigned)b), dim3(256), 0, stream>>>(x1, x2, x33, x4, trans, tf, (float*)d_out);
  }
}